// GATNet_69947837383283
// MI455X (gfx1250) — compile-verified
//
#include <hip/hip_runtime.h>
#include <hip/hip_bf16.h>

#define N_NODES   50000
#define N_EDGES   800000
#define IN_DIM    128
#define HID       16
#define HEADS     8
#define N_CLASSES 10
#define NEG_SLOPE 0.2f

typedef float v2f __attribute__((ext_vector_type(2)));
typedef float v8f __attribute__((ext_vector_type(8)));

// ---------------------------------------------------------------------------
// GEMM: C[M x NC] = A[M x 128] * B[128 x NC], one wave per 16x16 tile.
// M is a multiple of 16 (50000 = 3125*16). NC is a compile-time constant
// (128 for layers 0-3, 10 for layer 4) so all B offsets are immediates.
// For NC=10, lanes 10..15 read a clamped (in-bounds) B column and compute
// garbage in their own output column only (B rows are striped one column per
// lane), which is masked at the store -- no per-iteration predication needed.
// ---------------------------------------------------------------------------
template <int NC>
__global__ __launch_bounds__(32) void gat_wmma_gemm_k128(
    const float* __restrict__ A, const float* __restrict__ B,
    float* __restrict__ C) {
  const int lane = threadIdx.x;
  const int half = lane >> 4;   // A: 0 -> K=0,1 | 1 -> K=2,3 ; same for B rows
  const int l16  = lane & 15;
  const int row0 = blockIdx.x * 16;
  const int col0 = blockIdx.y * 16;
  const int colb = col0 + l16;
  const int colc = (NC % 16 == 0) ? colb : (colb < NC ? colb : NC - 1);

  const float* __restrict__ arow = A + (size_t)(row0 + l16) * IN_DIM + 2 * half;
  const float* __restrict__ bcol = B + (size_t)(2 * half) * NC + colc;

  v8f acc = {};
#pragma unroll
  for (int k0 = 0; k0 < IN_DIM; k0 += 4) {
    v2f a, b;
    a.x = arow[k0];               // A[row][k0+2*half]
    a.y = arow[k0 + 1];           // A[row][k0+2*half+1]
    b.x = bcol[k0 * NC];          // B[k0+2*half][col]      (immediate offset)
    b.y = bcol[(k0 + 1) * NC];    // B[k0+2*half+1][col]
    acc = __builtin_amdgcn_wmma_f32_16x16x4_f32(false, a, false, b,
                                                (short)0, acc, false, false);
  }
  if (NC % 16 == 0 || colb < NC) {
    const int rbase = row0 + 8 * half;  // C/D: lanes 16-31 hold M=8..15
#pragma unroll
    for (int r = 0; r < 8; ++r)
      C[(size_t)(rbase + r) * NC + colb] = acc[r];
  }
}

// ---------------------------------------------------------------------------
// el[n,h] = dot(feat[n,h,:], al[h,:]) ; er analogous. One thread per (n,h).
// ---------------------------------------------------------------------------
template <int H, int D>
__global__ void gat_el_er(const float* __restrict__ feat,
                          const float* __restrict__ al,
                          const float* __restrict__ ar,
                          float* __restrict__ elb, float* __restrict__ erb,
                          int NH) {
  int i = blockIdx.x * blockDim.x + threadIdx.x;
  if (i >= NH) return;
  const int h = i % H;
  const float* __restrict__ f = feat + (size_t)i * D;
  float sl = 0.f, sr = 0.f;
#pragma unroll
  for (int d = 0; d < D; ++d) {
    const float fv = f[d];
    sl = fmaf(fv, al[h * D + d], sl);
    sr = fmaf(fv, ar[h * D + d], sr);
  }
  elb[i] = sl;
  erb[i] = sr;
}

// Monotonic uint encoding of float for atomicMax-based segment_max.
__device__ __forceinline__ unsigned f32_key(float x) {
  unsigned b = __float_as_uint(x);
  return (b & 0x80000000u) ? ~b : (b | 0x80000000u);
}
__device__ __forceinline__ float key_f32(unsigned k) {
  return __uint_as_float((k & 0x80000000u) ? (k & 0x7FFFFFFFu) : ~k);
}

// Pass 1: e = leaky_relu(el[src]+er[dst]); store e; segment_max into emax[dst].
template <int H>
__global__ void gat_edge_max(const int* __restrict__ src,
                             const int* __restrict__ dst,
                             const float* __restrict__ elb,
                             const float* __restrict__ erb,
                             float* __restrict__ ebuf,
                             unsigned* __restrict__ emax, int EH) {
  int i = blockIdx.x * blockDim.x + threadIdx.x;
  if (i >= EH) return;
  const int e = i / H, h = i - e * H;
  const int s = src[e], d = dst[e];
  float x = elb[s * H + h] + erb[d * H + h];
  x = (x >= 0.f) ? x : NEG_SLOPE * x;
  ebuf[i] = x;
  atomicMax(&emax[d * H + h], f32_key(x));
}

// Pass 2: ex = exp(e - emax[dst]); denom[dst] += ex; store ex in place.
template <int H>
__global__ void gat_edge_exp(const int* __restrict__ dst,
                             const unsigned* __restrict__ emax,
                             float* __restrict__ ebuf,
                             float* __restrict__ denom, int EH) {
  int i = blockIdx.x * blockDim.x + threadIdx.x;
  if (i >= EH) return;
  const int e = i / H, h = i - e * H;
  const int d = dst[e];
  const float m = key_f32(emax[d * H + h]);
  const float ex = expf(ebuf[i] - m);
  ebuf[i] = ex;
  atomicAdd(&denom[d * H + h], ex);
}

// Pass 3: agg[dst,h,:] += feat[src,h,:] * ex. One thread per (edge, head).
// Loads hoisted ahead of the atomics so they fuse into global_load_b128.
template <int H, int D>
__global__ void gat_edge_scatter(const int* __restrict__ src,
                                 const int* __restrict__ dst,
                                 const float* __restrict__ feat,
                                 const float* __restrict__ ebuf,
                                 float* __restrict__ agg, int EH) {
  int i = blockIdx.x * blockDim.x + threadIdx.x;
  if (i >= EH) return;
  const int e = i / H, h = i - e * H;
  const float ex = ebuf[i];
  const int s = src[e], d = dst[e];
  const float* __restrict__ f = feat + ((size_t)s * H + h) * D;
  float* __restrict__       a = agg  + ((size_t)d * H + h) * D;
  float vals[D];
#pragma unroll
  for (int dd = 0; dd < D; ++dd) vals[dd] = f[dd] * ex;
#pragma unroll
  for (int dd = 0; dd < D; ++dd) atomicAdd(&a[dd], vals[dd]);
}

// Finalize: out = denom>0 ? agg/denom : 0 ; optional ELU.
template <int D>
__global__ void gat_finalize(const float* __restrict__ agg,
                             const float* __restrict__ denom,
                             float* __restrict__ outp, int NHD, int activate) {
  int i = blockIdx.x * blockDim.x + threadIdx.x;
  if (i >= NHD) return;
  const float den = denom[i / D];
  float v = (den > 0.f) ? agg[i] / den : 0.f;
  if (activate) v = (v > 0.f) ? v : (expf(v) - 1.f);
  outp[i] = v;
}

// ---------------------------------------------------------------------------
extern "C" void kernel_launch(void* const* d_in, const int* in_sizes, int n_in,
                              void* d_out, int out_size, void* d_ws,
                              size_t ws_size, hipStream_t stream) {
  const float* h0  = (const float*)d_in[0];
  const int*   src = (const int*)d_in[1];
  const int*   dst = (const int*)d_in[2];
  const float *W[5], *al[5], *ar[5];
  for (int i = 0; i < 5; ++i) {
    W[i]  = (const float*)d_in[3 + 3 * i];
    al[i] = (const float*)d_in[4 + 3 * i];
    ar[i] = (const float*)d_in[5 + 3 * i];
  }

  // Workspace layout: ~109 MB total.
  char* p = (char*)d_ws;
  float*    hbuf  = (float*)p;    p += (size_t)N_NODES * 128 * sizeof(float);
  float*    feat  = (float*)p;    p += (size_t)N_NODES * 128 * sizeof(float);
  float*    agg   = (float*)p;    p += (size_t)N_NODES * 128 * sizeof(float);
  float*    ebuf  = (float*)p;    p += (size_t)N_EDGES * HEADS * sizeof(float);
  float*    elb   = (float*)p;    p += (size_t)N_NODES * HEADS * sizeof(float);
  float*    erb   = (float*)p;    p += (size_t)N_NODES * HEADS * sizeof(float);
  unsigned* emax  = (unsigned*)p; p += (size_t)N_NODES * HEADS * sizeof(unsigned);
  float*    denom = (float*)p;

  for (int layer = 0; layer < 5; ++layer) {
    const bool last = (layer == 4);
    const int  H    = last ? 1 : HEADS;
    const int  HD   = last ? N_CLASSES : HEADS * HID;
    const float* Ain  = (layer == 0) ? h0 : hbuf;
    float*       outp = last ? (float*)d_out : hbuf;

    hipMemsetAsync(agg,   0, (size_t)N_NODES * HD * sizeof(float),   stream);
    hipMemsetAsync(denom, 0, (size_t)N_NODES * H * sizeof(float),    stream);
    hipMemsetAsync(emax,  0, (size_t)N_NODES * H * sizeof(unsigned), stream);

    // feat = Ain @ W[layer]  (50000x128 @ 128xHD) via fp32 WMMA 16x16x4
    if (!last) {
      dim3 g(N_NODES / 16, HEADS * HID / 16);
      gat_wmma_gemm_k128<HEADS * HID><<<g, 32, 0, stream>>>(Ain, W[layer], feat);
    } else {
      dim3 g(N_NODES / 16, 1);
      gat_wmma_gemm_k128<N_CLASSES><<<g, 32, 0, stream>>>(Ain, W[layer], feat);
    }

    int t = N_NODES * H;
    if (!last)
      gat_el_er<HEADS, HID><<<(t + 255) / 256, 256, 0, stream>>>(
          feat, al[layer], ar[layer], elb, erb, t);
    else
      gat_el_er<1, N_CLASSES><<<(t + 255) / 256, 256, 0, stream>>>(
          feat, al[layer], ar[layer], elb, erb, t);

    t = N_EDGES * H;
    const int eb = (t + 255) / 256;
    if (!last) {
      gat_edge_max<HEADS><<<eb, 256, 0, stream>>>(src, dst, elb, erb, ebuf,
                                                  emax, t);
      gat_edge_exp<HEADS><<<eb, 256, 0, stream>>>(dst, emax, ebuf, denom, t);
      gat_edge_scatter<HEADS, HID><<<eb, 256, 0, stream>>>(src, dst, feat,
                                                           ebuf, agg, t);
    } else {
      gat_edge_max<1><<<eb, 256, 0, stream>>>(src, dst, elb, erb, ebuf, emax, t);
      gat_edge_exp<1><<<eb, 256, 0, stream>>>(dst, emax, ebuf, denom, t);
      gat_edge_scatter<1, N_CLASSES><<<eb, 256, 0, stream>>>(src, dst, feat,
                                                             ebuf, agg, t);
    }

    t = N_NODES * HD;
    if (!last)
      gat_finalize<HID><<<(t + 255) / 256, 256, 0, stream>>>(agg, denom, outp,
                                                             t, 1);
    else
      gat_finalize<N_CLASSES><<<(t + 255) / 256, 256, 0, stream>>>(
          agg, denom, outp, t, 0);
  }
  // Final layer writes (N,10) directly to d_out; the reference's mean over the
  // singleton head axis is the identity.
}